// RepulsionEnergyFixed_2628519985580
// MI455X (gfx1250) — compile-verified
//
#include <hip/hip_runtime.h>
#include <math.h>
#include <stdint.h>

// ---------------------------------------------------------------------------
// RepulsionEnergy on MI455X (gfx1250, wave32).
//
// d2 tile trick: A row i = (-2x,-2y,-2z, |ri|^2), B col j = (x,y,z,1),
// C = splat(|rj|^2)  =>  D = A*B + C = full 16x16 d^2 tile from ONE
// v_wmma_f32_16x16x4_f32.
//
// Exact top-64 per row via 256-bin d^2 histogram in [0,100) + boundary-bin
// refinement (selection by value -> deterministic).
//
// CDNA5 paths: v_wmma_f32_16x16x4_f32, global_load_async_to_lds_b128 +
// s_wait_asynccnt staging, ds_add_u32 histograms, wave32 shuffles.
// ---------------------------------------------------------------------------

typedef float v2f __attribute__((ext_vector_type(2)));
typedef float v8f __attribute__((ext_vector_type(8)));

#define LPTS   2048
#define KSEL   64
#define NBINS  256
#define CAP    96          // boundary-bin candidate capacity per row
#define NWAVES 4
#define TILE   16
#define NCT    (LPTS / TILE)   // 128 column tiles
#define NRT    (LPTS / TILE)   // 128 row strips per batch
#define RCUT2  100.0f
#define BIN_SCALE ((float)NBINS / RCUT2)

#if __has_builtin(__builtin_amdgcn_global_load_async_to_lds_b128) && \
    __has_builtin(__builtin_amdgcn_s_wait_asynccnt)
#define USE_ASYNC_LDS 1
#else
#define USE_ASYNC_LDS 0
#endif

#if USE_ASYNC_LDS
// The builtin takes (AS1 int4vec*, AS3 int4vec*, imm offset, imm cpol);
// clang prints AS1 as "__device__" in HIP mode. Cast via uintptr_t (no
// implicit void*->T* conversion exists in C++).
typedef int asyncvec_t __attribute__((vector_size(16)));
typedef __attribute__((address_space(1))) asyncvec_t* g_asyncvec_p;
typedef __attribute__((address_space(3))) asyncvec_t* l_asyncvec_p;

__device__ __forceinline__ void async_copy_b128(const void* g, void* lds) {
    __builtin_amdgcn_global_load_async_to_lds_b128(
        (g_asyncvec_p)(uintptr_t)g,
        (l_asyncvec_p)(uint32_t)(uintptr_t)lds,
        0, 0);
}
#endif

// f(r) = WALL * softplus((R0 - r)/DELTA) * smoothstep(1->0 over [R_ON, R_CUT])
__device__ __forceinline__ float frep_d2(float d2) {
    float r  = sqrtf(fmaxf(d2, 1e-12f));
    float t  = fminf(fmaxf((r - 8.0f) * 0.5f, 0.0f), 1.0f);
    float sw = 1.0f - t * t * (3.0f - 2.0f * t);
    float x  = (4.0f - r) * 4.999999999975e0f;      // 1/(0.2 + 1e-12)
    float sp = (x > 20.0f) ? x : log1pf(expf(x));
    return 10.0f * sp * sw;
}

// One 16x16 d^2 tile via a single WMMA. Lane layout (ISA 7.12.2):
//   A (16x4, 32-bit): lane<16 -> row (lane&15) K=0,1 ; lane>=16 -> K=2,3
//   B (4x16, 32-bit): lane<16 -> col (lane&15) K=0,1 ; lane>=16 -> K=2,3
//   C/D: VGPR v, lane<16 -> M=v, N=lane ; lane>=16 -> M=v+8, N=lane-16
__device__ __forceinline__ v8f d2_tile(const float4* __restrict__ pts,
                                       v2f afrag, int colBase,
                                       int half, int l16) {
    float4 pB = pts[colBase + l16];
    v2f b;
    b.x = half ? pB.z : pB.x;
    b.y = half ? 1.0f : pB.y;
    float x2j = pB.w;                       // |r_j|^2, constant per column
    v8f c = {x2j, x2j, x2j, x2j, x2j, x2j, x2j, x2j};
    return __builtin_amdgcn_wmma_f32_16x16x4_f32(
        false, afrag, false, b, (short)0, c, false, false);
}

// EXCL=true only for the <=3 column tiles straddling the diagonal.
template <bool EXCL>
__device__ __forceinline__ void hist_tile(v8f d, int rowBase, int colBase,
                                          int half, int l16,
                                          unsigned (*hist)[NBINS]) {
#pragma unroll
    for (int v = 0; v < 8; ++v) {
        int row  = v + 8 * half;
        float d2 = fmaxf(d[v], 0.0f);
        bool ok  = d2 < RCUT2;
        if (EXCL) {
            int diff = (rowBase + row) - (colBase + l16);
            ok = ok && !(diff >= -1 && diff <= 1);
        }
        if (ok) {
            int bin = min((int)(d2 * BIN_SCALE), NBINS - 1);
            atomicAdd(&hist[row][bin], 1u);
        }
    }
}

template <bool EXCL>
__device__ __forceinline__ void accum_tile(v8f d, int rowBase, int colBase,
                                           int half, int l16, const int tl[8],
                                           float acc[8], unsigned* candCnt,
                                           float (*cand)[CAP]) {
#pragma unroll
    for (int v = 0; v < 8; ++v) {
        int row  = v + 8 * half;
        float d2 = fmaxf(d[v], 0.0f);
        bool ok  = d2 < RCUT2;
        if (EXCL) {
            int diff = (rowBase + row) - (colBase + l16);
            ok = ok && !(diff >= -1 && diff <= 1);
        }
        if (ok) {
            int bin = min((int)(d2 * BIN_SCALE), NBINS - 1);
            if (bin < tl[v]) {
                acc[v] += frep_d2(d2);
            } else if (bin == tl[v]) {
                unsigned s = atomicAdd(&candCnt[row], 1u);
                if (s < CAP) cand[row][s] = d2;
            }
        }
    }
}

__global__ __launch_bounds__(NWAVES * 32)
void repulsion_strip_kernel(const float* __restrict__ R,
                            float* __restrict__ strip_out) {
    __shared__ float4 pts[LPTS];                 // (x,y,z,|r|^2)      32 KB
    // staging buffer (load phase) aliases the selection state (compute phase)
    __shared__ union U {
        float stage[LPTS * 3];                   // 24 KB raw R staging
        struct {
            unsigned hist[TILE][NBINS];          // 16 KB
            float    cand[TILE][CAP];            //  6 KB
            float    partial[TILE][NWAVES];
            unsigned candCnt[TILE];
            int      Tbin[TILE];
            int      needT[TILE];
            float    rowTot[TILE];
        } sel;
    } u;

    const int tid  = threadIdx.x;
    const int wave = tid >> 5;
    const int lane = tid & 31;
    const int half = lane >> 4;     // which 16-lane half of the wave
    const int l16  = lane & 15;

    const int b       = blockIdx.x / NRT;
    const int rowBase = (blockIdx.x % NRT) * TILE;
    const float* Rb   = R + (size_t)b * LPTS * 3;

    // ---- stage coordinates into LDS, then build (x,y,z,|r|^2) table ----
#if USE_ASYNC_LDS
    {
        const float4* gsrc = (const float4*)Rb;          // 24576 B, 16B-aligned
        float4* ldst       = (float4*)u.stage;
        for (int c = tid; c < (LPTS * 3) / 4; c += blockDim.x)
            async_copy_b128(&gsrc[c], &ldst[c]);
        __builtin_amdgcn_s_wait_asynccnt(0);             // my copies landed
    }
    __syncthreads();                                     // everyone's landed
    for (int p = tid; p < LPTS; p += blockDim.x) {
        float x = u.stage[p * 3 + 0];
        float y = u.stage[p * 3 + 1];
        float z = u.stage[p * 3 + 2];
        pts[p] = make_float4(x, y, z, x * x + y * y + z * z);
    }
    __syncthreads();                                     // stage consumed
#else
    for (int p = tid; p < LPTS; p += blockDim.x) {
        float x = Rb[p * 3 + 0], y = Rb[p * 3 + 1], z = Rb[p * 3 + 2];
        pts[p] = make_float4(x, y, z, x * x + y * y + z * z);
    }
#endif

    // ---- zero selection state (overwrites staging union member) ----
    for (int i = tid; i < TILE * NBINS; i += blockDim.x)
        ((unsigned*)u.sel.hist)[i] = 0u;
    if (tid < TILE * NWAVES) ((float*)u.sel.partial)[tid] = 0.0f;
    if (tid < TILE)          u.sel.candCnt[tid] = 0u;
    __syncthreads();

    // A fragment for this row strip (branch-free lane select, keeps EXEC full)
    float4 pA = pts[rowBase + l16];
    v2f afrag;
    afrag.x = half ? (-2.0f * pA.z) : (-2.0f * pA.x);
    afrag.y = half ?  pA.w          : (-2.0f * pA.y);

    // ---- pass 1: per-row d^2 histogram over [0, 100) ----
    for (int ct = wave; ct < NCT; ct += NWAVES) {
        const int colBase = ct * TILE;
        v8f d = d2_tile(pts, afrag, colBase, half, l16);
        const int dd = colBase - rowBase;                // wave-uniform branch
        if (dd >= -TILE && dd <= TILE)
            hist_tile<true >(d, rowBase, colBase, half, l16, u.sel.hist);
        else
            hist_tile<false>(d, rowBase, colBase, half, l16, u.sel.hist);
    }
    __syncthreads();

    // ---- per-row threshold bin: first bin where cumcount reaches 64 ----
    if (tid < TILE) {
        unsigned cum = 0; int T = NBINS, need = 0;
        for (int bin = 0; bin < NBINS; ++bin) {
            unsigned h = u.sel.hist[tid][bin];
            if (T == NBINS && cum + h >= (unsigned)KSEL) {
                T = bin; need = KSEL - (int)cum;
            }
            cum += h;
        }
        if (cum <= (unsigned)KSEL) { T = NBINS; need = 0; } // take everything
        u.sel.Tbin[tid] = T; u.sel.needT[tid] = need;
    }
    __syncthreads();

    int tl[8];
#pragma unroll
    for (int v = 0; v < 8; ++v) tl[v] = u.sel.Tbin[v + 8 * half];

    // ---- pass 2: accumulate f(r) below threshold, stash boundary bin ----
    float acc[8] = {0.f, 0.f, 0.f, 0.f, 0.f, 0.f, 0.f, 0.f};
    for (int ct = wave; ct < NCT; ct += NWAVES) {
        const int colBase = ct * TILE;
        v8f d = d2_tile(pts, afrag, colBase, half, l16);
        const int dd = colBase - rowBase;                // wave-uniform branch
        if (dd >= -TILE && dd <= TILE)
            accum_tile<true >(d, rowBase, colBase, half, l16, tl, acc,
                              u.sel.candCnt, u.sel.cand);
        else
            accum_tile<false>(d, rowBase, colBase, half, l16, tl, acc,
                              u.sel.candCnt, u.sel.cand);
    }
    // deterministic cross-lane reduce within each 16-lane half (one row each)
#pragma unroll
    for (int v = 0; v < 8; ++v) {
        float a = acc[v];
        for (int m = 8; m >= 1; m >>= 1) a += __shfl_xor(a, m, 16);
        if (l16 == 0) u.sel.partial[v + 8 * half][wave] = a;
    }
    __syncthreads();

    // ---- boundary bin: exact m-smallest by value, then row totals ----
    if (tid < TILE) {
        int   m = u.sel.needT[tid];
        int   c = (int)min(u.sel.candCnt[tid], (unsigned)CAP);
        float s = 0.0f;
        for (int k = 0; k < m && k < c; ++k) {
            int best = -1; float bv = 3.4e38f;
            for (int q = 0; q < c; ++q) {
                float vq = u.sel.cand[tid][q];
                if (vq < bv) { bv = vq; best = q; }
            }
            if (best >= 0) { s += frep_d2(bv); u.sel.cand[tid][best] = 3.4e38f; }
        }
        for (int w = 0; w < NWAVES; ++w) s += u.sel.partial[tid][w];
        u.sel.rowTot[tid] = s;
    }
    __syncthreads();
    if (tid == 0) {
        float s = 0.0f;
        for (int r = 0; r < TILE; ++r) s += u.sel.rowTot[r];
        strip_out[blockIdx.x] = s;
    }
}

// fixed-tree reduction of 128 strip sums per batch -> deterministic output
__global__ __launch_bounds__(NRT)
void reduce_strips_kernel(const float* __restrict__ strips,
                          float* __restrict__ out) {
    __shared__ float sm[NRT];
    sm[threadIdx.x] = strips[blockIdx.x * NRT + threadIdx.x];
    __syncthreads();
    for (int s = NRT / 2; s > 0; s >>= 1) {
        if ((int)threadIdx.x < s) sm[threadIdx.x] += sm[threadIdx.x + s];
        __syncthreads();
    }
    if (threadIdx.x == 0) out[blockIdx.x] = sm[0];
}

extern "C" void kernel_launch(void* const* d_in, const int* in_sizes, int n_in,
                              void* d_out, int out_size, void* d_ws, size_t ws_size,
                              hipStream_t stream) {
    const float* R   = (const float*)d_in[0];
    float* out       = (float*)d_out;
    float* strips    = (float*)d_ws;                  // B * 128 floats
    const int B      = in_sizes[0] / (LPTS * 3);      // = 8

    repulsion_strip_kernel<<<B * NRT, NWAVES * 32, 0, stream>>>(R, strips);
    reduce_strips_kernel<<<B, NRT, 0, stream>>>(strips, out);
}